// MemoryBank_83356725281406
// MI455X (gfx1250) — compile-verified
//
#include <hip/hip_runtime.h>
#include <hip/hip_bf16.h>
#include <math.h>

typedef __attribute__((ext_vector_type(2))) float v2f;
typedef __attribute__((ext_vector_type(4))) float v4f;
typedef __attribute__((ext_vector_type(8))) float v8f;

#define B_   16
#define S_   2048
#define D_   1024
#define M_   512
#define T_   256
#define TOPK 8

// ---------------------------------------------------------------------------
// Kernel 1: router_scores[16x512] = x0[16x1024] @ router[1024x512]
// via V_WMMA_F32_16X16X4_F32. One wave per 16-col tile, 32 tiles.
// A layout (16x4 f32): lane&15 = M; K = 2*(lane>>4) + {vgpr0,vgpr1}
// B layout (4x16 f32): lane&15 = N; K = 2*(lane>>4) + {vgpr0,vgpr1}
// C/D layout: VGPR v -> M = v + (lane<16 ? 0 : 8), N = lane&15
// ---------------------------------------------------------------------------
__global__ __launch_bounds__(32)
void router_wmma_kernel(const float* __restrict__ x,
                        const float* __restrict__ router,
                        float* __restrict__ scores) {
  const int lane  = threadIdx.x;       // 0..31, full wave (EXEC all ones)
  const int n0    = blockIdx.x * 16;   // output column tile base
  const int row   = lane & 15;         // A-row (batch) / B-col offset
  const int khalf = (lane >> 4) * 2;   // 0 or 2

  const float* xrow = x + (size_t)row * S_ * D_;   // x[row, 0, :]

  v8f c = {};
  for (int k = 0; k < D_; k += 4) {
    v2f a, b;
    a.x = xrow[k + khalf];
    a.y = xrow[k + khalf + 1];
    b.x = router[(size_t)(k + khalf)     * M_ + n0 + row];
    b.y = router[(size_t)(k + khalf + 1) * M_ + n0 + row];
    // 8 args: (neg_a, A, neg_b, B, c_mod, C, reuse_a, reuse_b)
    c = __builtin_amdgcn_wmma_f32_16x16x4_f32(false, a, false, b,
                                              (short)0, c, false, false);
  }

  const int mbase = (lane < 16) ? 0 : 8;
  const int col   = n0 + row;
#pragma unroll
  for (int v = 0; v < 8; ++v) {
    scores[(size_t)(v + mbase) * M_ + col] = c[v];
  }
}

// ---------------------------------------------------------------------------
// Kernel 2: per-batch top-8 + softmax. One wave per batch (16 blocks x 32).
// Each lane owns 16 of the 512 scores; 8 rounds of wave-argmax via shuffles.
// ---------------------------------------------------------------------------
__global__ __launch_bounds__(32)
void topk_softmax_kernel(const float* __restrict__ scores,
                         int* __restrict__ topi,
                         float* __restrict__ topw) {
  const int b    = blockIdx.x;
  const int lane = threadIdx.x;

  float loc[16];
  const float* srow = scores + (size_t)b * M_;
#pragma unroll
  for (int j = 0; j < 16; ++j) loc[j] = srow[lane * 16 + j];

  float tv[TOPK];
  int   ti[TOPK];
  for (int r = 0; r < TOPK; ++r) {
    // local argmax over this lane's 16 entries
    float val = -INFINITY;
    int   li  = 0x7fffffff;
#pragma unroll
    for (int j = 0; j < 16; ++j) {
      if (loc[j] > val) { val = loc[j]; li = lane * 16 + j; }
    }
    // wave reduction (lowest index wins ties -> matches lax.top_k)
#pragma unroll
    for (int off = 16; off >= 1; off >>= 1) {
      float ov = __shfl_xor(val, off, 32);
      int   oi = __shfl_xor(li,  off, 32);
      if (ov > val || (ov == val && oi < li)) { val = ov; li = oi; }
    }
    tv[r] = val;
    ti[r] = li;
    // owning lane retires the winner (static unroll -> stays in registers)
    if ((li >> 4) == lane) {
      const int jj = li & 15;
#pragma unroll
      for (int j = 0; j < 16; ++j)
        if (j == jj) loc[j] = -INFINITY;
    }
  }

  if (lane == 0) {
    const float mx = tv[0];          // rounds are descending -> tv[0] is max
    float e[TOPK], sum = 0.f;
#pragma unroll
    for (int r = 0; r < TOPK; ++r) { e[r] = __expf(tv[r] - mx); sum += e[r]; }
    const float inv = 1.f / sum;
#pragma unroll
    for (int r = 0; r < TOPK; ++r) {
      topi[b * TOPK + r] = ti[r];
      topw[b * TOPK + r] = e[r] * inv;
    }
  }
}

// ---------------------------------------------------------------------------
// Kernel 3: fused combine + copy. One block per (b, t) row of the output;
// 256 threads x float4 = 1024 floats = one row. b,t are block-uniform, so
// the weight/index loads are scalar. Streaming paths use non-temporal hints.
// ---------------------------------------------------------------------------
__global__ __launch_bounds__(256)
void combine_copy_kernel(const float* __restrict__ x,
                         const float* __restrict__ mem,
                         const int* __restrict__ topi,
                         const float* __restrict__ topw,
                         float* __restrict__ out) {
  const int blk = blockIdx.x;          // 0 .. 16*2048-1
  const int t   = blk & (S_ - 1);
  const int b   = blk >> 11;           // / 2048
  const int d4  = threadIdx.x;         // float4 index within the row

  const size_t rowoff = ((size_t)b * S_ + t) * D_ + (size_t)d4 * 4;
  v4f* o = (v4f*)(out + rowoff);

  if (t == 0 || t > T_) {
    // pure copy region: stream x -> out, bypass cache retention
    const v4f* src = (const v4f*)(x + rowoff);
    v4f v = __builtin_nontemporal_load(src);
    __builtin_nontemporal_store(v, o);
  } else {
    const int tt = t - 1;
    v4f acc = {0.f, 0.f, 0.f, 0.f};
#pragma unroll
    for (int k = 0; k < TOPK; ++k) {
      const float w = topw[b * TOPK + k];   // block-uniform -> s_load
      const int   m = topi[b * TOPK + k];
      const v4f v = *(const v4f*)(mem + ((size_t)m * T_ + tt) * D_ + (size_t)d4 * 4);
      acc += w * v;   // ext-vector arithmetic -> v_fma on 4 components
    }
    __builtin_nontemporal_store(acc, o);
  }
}

// ---------------------------------------------------------------------------
extern "C" void kernel_launch(void* const* d_in, const int* in_sizes, int n_in,
                              void* d_out, int out_size, void* d_ws, size_t ws_size,
                              hipStream_t stream) {
  const float* x      = (const float*)d_in[0];   // (16, 2048, 1024) f32
  const float* mem    = (const float*)d_in[1];   // (512, 256, 1024) f32
  const float* router = (const float*)d_in[2];   // (1024, 512) f32
  float*       out    = (float*)d_out;           // (16, 2048, 1024) f32

  float* scores = (float*)d_ws;                  // 16*512 f32
  int*   topi   = (int*)(scores + B_ * M_);      // 16*8 i32
  float* topw   = (float*)(topi + B_ * TOPK);    // 16*8 f32

  router_wmma_kernel<<<M_ / 16, 32, 0, stream>>>(x, router, scores);
  topk_softmax_kernel<<<B_, 32, 0, stream>>>(scores, topi, topw);
  combine_copy_kernel<<<B_ * S_, 256, 0, stream>>>(x, mem, topi, topw, out);
}